// LocalAttention2d_57621281243441
// MI455X (gfx1250) — compile-verified
//
#include <hip/hip_runtime.h>
#include <math.h>

typedef __attribute__((ext_vector_type(2))) float v2f;
typedef __attribute__((ext_vector_type(8))) float v8f;

#define Bq   64
#define Dq   128
#define Hq   128
#define Wq   128
#define CSZ  256
#define ROWS 129   // Hq + 1 (NaN-border padded)

// ---------------------------------------------------------------------------
// Kernel 1: v = c_t @ W_a   (M=64, K=256, N=128), fp32 WMMA 16x16x4.
// One wave per 16x16 output tile; 32 waves total = 4 blocks x 8 waves.
// A-frag (16x4 f32): lanes 0-15 -> M=lane, K pair {k0,k0+1};
//                    lanes16-31 -> M=lane-16, K pair {k0+2,k0+3}.
// B-frag (4x16 f32): VGPR0 = rows K=k0 (lanes 0-15) / K=k0+2 (lanes 16-31),
//                    VGPR1 = rows K=k0+1 / k0+3; N = lane & 15.
// D (16x16 f32):     VGPR g -> row g + 8*(lane>>4), col = lane & 15.
// ---------------------------------------------------------------------------
__global__ __launch_bounds__(256)
void la2d_gemm_v_kernel(const float* __restrict__ c_t,
                        const float* __restrict__ W_a,
                        float* __restrict__ v_out) {
  const int lane = threadIdx.x & 31;
  const int wave = blockIdx.x * (blockDim.x >> 5) + (threadIdx.x >> 5); // 0..31
  const int mt = wave >> 3;   // 0..3
  const int nt = wave & 7;    // 0..7
  const int l  = lane & 15;
  const int hi = lane >> 4;   // 0 or 1

  const int arow = mt * 16 + l;   // c_t row feeding A
  const int bcol = nt * 16 + l;   // W_a column feeding B

  v8f acc = {0.f, 0.f, 0.f, 0.f, 0.f, 0.f, 0.f, 0.f};
  for (int k0 = 0; k0 < CSZ; k0 += 4) {
    const int ka = k0 + 2 * hi;
    v2f a, b;
    a.x = c_t[arow * CSZ + ka];
    a.y = c_t[arow * CSZ + ka + 1];
    b.x = W_a[(ka    ) * Dq + bcol];
    b.y = W_a[(ka + 1) * Dq + bcol];
    acc = __builtin_amdgcn_wmma_f32_16x16x4_f32(
        /*neg_a=*/false, a, /*neg_b=*/false, b,
        /*c_mod=*/(short)0, acc, /*reuse_a=*/false, /*reuse_b=*/false);
  }

#pragma unroll
  for (int g = 0; g < 8; ++g) {
    const int row = mt * 16 + g + 8 * hi;
    v_out[row * Dq + bcol] = acc[g];
  }
}

// ---------------------------------------------------------------------------
// Kernel 2: per-batch window attention. One 256-thread block per batch.
// ---------------------------------------------------------------------------
__global__ __launch_bounds__(256)
void la2d_attn_kernel(const float* __restrict__ q,
                      const float* __restrict__ c_t,
                      const float* __restrict__ W_p,
                      const float* __restrict__ v_in,
                      float* __restrict__ out) {
  const int b = blockIdx.x;
  const int t = threadIdx.x;

  __shared__ float sQ[64 * 130];  // qg, padded stride to dodge LDS bank conflicts
  __shared__ float sV[Dq];
  __shared__ float sA[64];
  __shared__ float sW[64];
  __shared__ float sPt[2];
  __shared__ int   sR[8], sC[8];
  __shared__ float sRe[8], sCe[8];
  __shared__ float sMS[2];        // softmax max, sum

  // p_t = Hq * sigmoid(c_t @ W_p^T)
  if (t < 2) {
    const float* wp = W_p + t * CSZ;
    const float* ct = c_t + b * CSZ;
    float dot = 0.f;
    for (int k = 0; k < CSZ; ++k) dot += ct[k] * wp[k];
    sPt[t] = (float)Hq / (1.f + expf(-dot));
  }
  __syncthreads();

  // window indices + gaussian shift terms (bug-faithful: cols use rows=129 too)
  if (t < 8) {
    const int off = t - 3;                 // arange(8) + (1 - R/2)
    const float p0 = sPt[0], p1 = sPt[1];
    const int pr0 = (int)rintf(p0);        // round-half-to-even
    const int pr1 = (int)rintf(p1);
    int rr = pr0 + off; rr = rr < 0 ? 0 : (rr > ROWS ? ROWS : rr); if (rr == ROWS) rr = 0;
    int cc = pr1 + off; cc = cc < 0 ? 0 : (cc > ROWS ? ROWS : cc); if (cc == ROWS) cc = 0;
    sR[t] = rr; sC[t] = cc;
    const float fr = (float)(rr - 1 > 0 ? rr - 1 : 0);
    const float fc = (float)(cc - 1 > 0 ? cc - 1 : 0);
    const float dr = (fr - p0) * 0.25f;    // /(R/2)
    const float dc = (fc - p1) * 0.25f;    // /(C/2)
    sRe[t] = 2.f * dr * dr;
    sCe[t] = 2.f * dc * dc;
  }
  if (t < Dq) sV[t] = v_in[b * Dq + t];    // v = W_a^T c_t, from kernel 1
  __syncthreads();

  // gather 8x8 window x 128 channels into LDS (k fastest across lanes)
  const float* qb = q + (size_t)b * Dq * Hq * Wq;
  for (int e = t; e < 64 * Dq; e += 256) {
    const int k = e & 63;
    const int d = e >> 6;
    const int i = k >> 3, j = k & 7;
    const int rr = sR[i], cc = sC[j];
    float x = 0.f;
    if (rr > 0 && cc > 0) {
      x = qb[((size_t)d * Hq + (rr - 1)) * Wq + (cc - 1)];
    }
    sQ[k * 130 + d] = x;
  }
  __syncthreads();

  // scores: a[k] = qg[k] . v  -  shift[k]  (invalid -> -inf)
  if (t < 64) {
    float sum = 0.f;
    for (int d = 0; d < Dq; ++d) sum += sQ[t * 130 + d] * sV[d];
    const int i = t >> 3, j = t & 7;
    const bool val = (sR[i] > 0) && (sC[j] > 0);
    sA[t] = val ? (sum - (sRe[i] + sCe[j])) : -INFINITY;
  }
  __syncthreads();

  if (t == 0) {
    float amax = -INFINITY;
    for (int k = 0; k < 64; ++k) amax = fmaxf(amax, sA[k]);
    float s = 0.f;
    for (int k = 0; k < 64; ++k) s += expf(sA[k] - amax);
    sMS[0] = amax; sMS[1] = s;
  }
  __syncthreads();

  if (t < 64) sW[t] = expf(sA[t] - sMS[0]) / sMS[1];
  __syncthreads();

  // out[d] = sum_k w[k] * qg[k][d]
  if (t < Dq) {
    float o = 0.f;
    for (int k = 0; k < 64; ++k) o += sW[k] * sQ[k * 130 + t];
    out[b * Dq + t] = o;
  }
}

extern "C" void kernel_launch(void* const* d_in, const int* in_sizes, int n_in,
                              void* d_out, int out_size, void* d_ws, size_t ws_size,
                              hipStream_t stream) {
  const float* q   = (const float*)d_in[0];   // [64,128,128,128]
  const float* c_t = (const float*)d_in[1];   // [64,256]
  const float* W_a = (const float*)d_in[2];   // [256,128]
  const float* W_p = (const float*)d_in[3];   // [2,256]
  float* out  = (float*)d_out;                // [64,128]
  float* v_ws = (float*)d_ws;                 // 64*128 floats scratch

  la2d_gemm_v_kernel<<<4, 256, 0, stream>>>(c_t, W_a, v_ws);
  la2d_attn_kernel<<<Bq, 256, 0, stream>>>(q, c_t, W_p, v_ws, out);
}